// SPGPromptEncoder_78469052498351
// MI455X (gfx1250) — compile-verified
//
#include <hip/hip_runtime.h>
#include <hip/hip_bf16.h>
#include <math.h>

typedef __attribute__((ext_vector_type(2))) float v2f;
typedef __attribute__((ext_vector_type(8))) float v8f;

#define BATCH 64
#define IMG   256
#define NPIX  (IMG*IMG)      // 65536
#define EMBED 256
#define H1    128
#define NP1   (H1*H1)        // 16384
#define H2    64
#define NP2   (H2*H2)        // 4096
#define C1    4
#define C2    16

// ---------------- Kernel 1: diff map + per-image mean/std threshold -------------
// image/mirror are read exactly once -> non-temporal loads (don't pollute L2).
__global__ void k_diff_stats(const float* __restrict__ img, const float* __restrict__ mir,
                             float* __restrict__ diff, float* __restrict__ thresh) {
  const int b   = blockIdx.x;
  const int tid = threadIdx.x;                 // 1024 threads
  const float* ib = img + (size_t)b * 3 * NPIX;
  const float* mb = mir + (size_t)b * 3 * NPIX;
  float* db = diff + (size_t)b * NPIX;
  float s = 0.f, sq = 0.f;
  for (int p = tid; p < NPIX; p += blockDim.x) {
    float i0 = __builtin_nontemporal_load(ib + p);
    float i1 = __builtin_nontemporal_load(ib + p + NPIX);
    float i2 = __builtin_nontemporal_load(ib + p + 2*NPIX);
    float m0 = __builtin_nontemporal_load(mb + p);
    float m1 = __builtin_nontemporal_load(mb + p + NPIX);
    float m2 = __builtin_nontemporal_load(mb + p + 2*NPIX);
    float d  = (fabsf(i0 - m0) + fabsf(i1 - m1) + fabsf(i2 - m2)) * (1.0f / 3.0f);
    db[p] = d;                                 // re-read twice -> keep temporal
    s += d; sq += d * d;
  }
  __shared__ float ss[1024], ssq[1024];
  ss[tid] = s; ssq[tid] = sq; __syncthreads();
  for (int o = 512; o > 0; o >>= 1) {
    if (tid < o) { ss[tid] += ss[tid + o]; ssq[tid] += ssq[tid + o]; }
    __syncthreads();
  }
  if (tid == 0) {
    float m   = ss[0] / (float)NPIX;
    float var = (ssq[0] - (float)NPIX * m * m) / (float)(NPIX - 1);
    var = fmaxf(var, 0.0f);
    thresh[b] = m + sqrtf(var);
  }
}

// ---------------- Kernel 2: centroid + sparse prompt embeddings -----------------
__global__ void k_centroid_sparse(const float* __restrict__ diff, const float* __restrict__ thresh,
                                  const float* __restrict__ gauss,   // (2,128)
                                  const float* __restrict__ nap,     // (256)
                                  const float* __restrict__ pos,     // (256)
                                  float* __restrict__ sparse) {      // (B,2,256)
  const int b   = blockIdx.x;
  const int tid = threadIdx.x;                 // 256 threads
  const float* db = diff + (size_t)b * NPIX;
  const float th  = thresh[b];
  float cnt = 0.f, xs = 0.f, ys = 0.f;
  for (int p = tid; p < NPIX; p += blockDim.x) {
    float m = (db[p] > th) ? 1.0f : 0.0f;
    cnt += m;
    xs  += m * (float)(p & (IMG - 1));
    ys  += m * (float)(p >> 8);
  }
  __shared__ float sc[256], sx[256], sy[256];
  sc[tid] = cnt; sx[tid] = xs; sy[tid] = ys; __syncthreads();
  for (int o = 128; o > 0; o >>= 1) {
    if (tid < o) { sc[tid] += sc[tid+o]; sx[tid] += sx[tid+o]; sy[tid] += sy[tid+o]; }
    __syncthreads();
  }
  __shared__ float scx, scy;
  if (tid == 0) {
    float c     = sc[0];
    float denom = fmaxf(c, 1.0f);
    float cx = sx[0] / denom, cy = sy[0] / denom;
    if (c == 0.0f) { cx = IMG * 0.5f; cy = IMG * 0.5f; }
    scx = cx; scy = cy;
  }
  __syncthreads();
  if (tid < 128) {
    float ccx = 2.0f * ((scx + 0.5f) / (float)IMG) - 1.0f;
    float ccy = 2.0f * ((scy + 0.5f) / (float)IMG) - 1.0f;
    float t = 6.283185307179586f * (ccx * gauss[tid] + ccy * gauss[128 + tid]);
    float* sp = sparse + (size_t)b * 2 * EMBED;
    sp[tid]               = sinf(t) + pos[tid];
    sp[128 + tid]         = cosf(t) + pos[128 + tid];
    sp[EMBED + tid]       = nap[tid];
    sp[EMBED + 128 + tid] = nap[128 + tid];
  }
}

__device__ __forceinline__ float gelu_exact(float v) {
  return 0.5f * v * (1.0f + erff(v * 0.7071067811865475f));
}

// ---------------- Kernel 3: conv1(2x2,s2,1->4) + channel-LN + GELU --------------
__global__ void k_conv1(const float* __restrict__ diff, const float* __restrict__ thresh,
                        const float* __restrict__ w1, const float* __restrict__ b1,
                        const float* __restrict__ lw1, const float* __restrict__ lb1,
                        float* __restrict__ x1) {                    // (B,4,128,128)
  int idx = blockIdx.x * blockDim.x + threadIdx.x;
  if (idx >= BATCH * NP1) return;
  int b  = idx / NP1;
  int p  = idx % NP1;
  int oy = p / H1, ox = p % H1;
  const float* db = diff + (size_t)b * NPIX;
  float th  = thresh[b];
  float m00 = (db[(2*oy    )*IMG + 2*ox    ] > th) ? 1.f : 0.f;
  float m01 = (db[(2*oy    )*IMG + 2*ox + 1] > th) ? 1.f : 0.f;
  float m10 = (db[(2*oy + 1)*IMG + 2*ox    ] > th) ? 1.f : 0.f;
  float m11 = (db[(2*oy + 1)*IMG + 2*ox + 1] > th) ? 1.f : 0.f;
  float y[C1];
  float u = 0.f;
  #pragma unroll
  for (int c = 0; c < C1; c++) {
    const float* w = w1 + c * 4;               // (4,1,2,2)
    float v = w[0]*m00 + w[1]*m01 + w[2]*m10 + w[3]*m11 + b1[c];
    y[c] = v; u += v;
  }
  u *= (1.0f / C1);
  float s = 0.f;
  #pragma unroll
  for (int c = 0; c < C1; c++) { float d = y[c] - u; s += d * d; }
  s *= (1.0f / C1);
  float inv = rsqrtf(s + 1e-6f);
  float* xb = x1 + (size_t)b * C1 * NP1;
  #pragma unroll
  for (int c = 0; c < C1; c++) {
    float v = (y[c] - u) * inv * lw1[c] + lb1[c];
    xb[(size_t)c * NP1 + p] = gelu_exact(v);
  }
}

// ---------------- Kernel 4: conv2(2x2,s2,4->16) + channel-LN + GELU -------------
__global__ void k_conv2(const float* __restrict__ x1,
                        const float* __restrict__ w2, const float* __restrict__ b2,
                        const float* __restrict__ lw2, const float* __restrict__ lb2,
                        float* __restrict__ x2) {                    // (B,16,64,64)
  int idx = blockIdx.x * blockDim.x + threadIdx.x;
  if (idx >= BATCH * NP2) return;
  int b  = idx / NP2;
  int p  = idx % NP2;
  int oy = p / H2, ox = p % H2;
  const float* xb = x1 + (size_t)b * C1 * NP1;
  float in[C1][4];
  #pragma unroll
  for (int ic = 0; ic < C1; ic++) {
    const float* xc = xb + (size_t)ic * NP1;
    in[ic][0] = xc[(2*oy    )*H1 + 2*ox    ];
    in[ic][1] = xc[(2*oy    )*H1 + 2*ox + 1];
    in[ic][2] = xc[(2*oy + 1)*H1 + 2*ox    ];
    in[ic][3] = xc[(2*oy + 1)*H1 + 2*ox + 1];
  }
  float y[C2];
  float u = 0.f;
  #pragma unroll
  for (int oc = 0; oc < C2; oc++) {
    float v = b2[oc];
    const float* w = w2 + oc * C1 * 4;         // (16,4,2,2)
    #pragma unroll
    for (int ic = 0; ic < C1; ic++)
      #pragma unroll
      for (int k = 0; k < 4; k++)
        v += w[ic*4 + k] * in[ic][k];
    y[oc] = v; u += v;
  }
  u *= (1.0f / C2);
  float s = 0.f;
  #pragma unroll
  for (int oc = 0; oc < C2; oc++) { float d = y[oc] - u; s += d * d; }
  s *= (1.0f / C2);
  float inv = rsqrtf(s + 1e-6f);
  float* ob = x2 + (size_t)b * C2 * NP2;
  #pragma unroll
  for (int oc = 0; oc < C2; oc++) {
    float v = (y[oc] - u) * inv * lw2[oc] + lb2[oc];
    ob[(size_t)oc * NP2 + p] = gelu_exact(v);
  }
}

// ---------------- Kernel 5: conv3 1x1 (16->256) via V_WMMA_F32_16X16X4_F32 ------
// Per wave: D(16 oc x 16 px) = A(16 oc x 16 ic) * B(16 ic x 16 px) + bias,
// done as four K=4 WMMA steps. M=oc on lanes->rows, N=px across lanes so the
// 8 result VGPRs store coalesced along pixels. dense (268 MB) exceeds the
// 192 MB L2 and is never re-read -> non-temporal streaming stores.
__global__ void k_conv3_wmma(const float* __restrict__ x2,   // (B,16,4096)
                             const float* __restrict__ w3,   // (256,16)
                             const float* __restrict__ b3,   // (256)
                             float* __restrict__ dense) {    // (B,256,4096)
  const int wave = (blockIdx.x * blockDim.x + threadIdx.x) >> 5;
  const int lane = threadIdx.x & 31;
  // 16 oc-tiles * 256 pixel-tiles = 4096 tiles per image
  const int b       = wave >> 12;
  const int t       = wave & 4095;
  const int oc_base = (t >> 8) << 4;           // 0..240
  const int p_base  = (t & 255) << 4;          // 0..4080
  const int half = lane >> 4;                  // K-half selector
  const int idx  = lane & 15;                  // M (for A) / N (for B,D)

  const float* xb = x2 + (size_t)b * C2 * NP2;

  v8f acc;
  #pragma unroll
  for (int r = 0; r < 8; r++) acc[r] = b3[oc_base + r + 8 * half];

  #pragma unroll
  for (int s = 0; s < 4; s++) {
    const int k0 = 4 * s;
    v2f a, bm;
    // A (16x4 f32): lanes 0-15 hold K=k0,k0+1 ; lanes 16-31 hold K=k0+2,k0+3
    a.x = w3[(oc_base + idx) * C2 + k0 + 2 * half];
    a.y = w3[(oc_base + idx) * C2 + k0 + 2 * half + 1];
    // B (4x16 f32): mirrored K striping, N across lanes
    bm.x = xb[(size_t)(k0 + 2 * half    ) * NP2 + p_base + idx];
    bm.y = xb[(size_t)(k0 + 2 * half + 1) * NP2 + p_base + idx];
    acc = __builtin_amdgcn_wmma_f32_16x16x4_f32(
        /*neg_a=*/false, a, /*neg_b=*/false, bm,
        /*c_mod=*/(short)0, acc, /*reuse_a=*/false, /*reuse_b=*/false);
  }

  float* ob = dense + (size_t)b * EMBED * NP2;
  #pragma unroll
  for (int r = 0; r < 8; r++)
    __builtin_nontemporal_store(acc[r],
        ob + (size_t)(oc_base + r + 8 * half) * NP2 + p_base + idx);
}

// --------------------------------- launcher -------------------------------------
extern "C" void kernel_launch(void* const* d_in, const int* in_sizes, int n_in,
                              void* d_out, int out_size, void* d_ws, size_t ws_size,
                              hipStream_t stream) {
  const float* image  = (const float*)d_in[0];
  const float* mirror = (const float*)d_in[1];
  const float* gauss  = (const float*)d_in[2];
  const float* nap    = (const float*)d_in[3];
  // d_in[4] = point_embed_neg (unused: labels are only {1,-1})
  const float* pos    = (const float*)d_in[5];
  const float* w1     = (const float*)d_in[6];
  const float* b1     = (const float*)d_in[7];
  const float* lw1    = (const float*)d_in[8];
  const float* lb1    = (const float*)d_in[9];
  const float* w2     = (const float*)d_in[10];
  const float* b2     = (const float*)d_in[11];
  const float* lw2    = (const float*)d_in[12];
  const float* lb2    = (const float*)d_in[13];
  const float* w3     = (const float*)d_in[14];
  const float* b3     = (const float*)d_in[15];

  float* ws      = (float*)d_ws;
  float* diff    = ws;                                       // B*NPIX
  float* thresh  = diff + (size_t)BATCH * NPIX;              // B
  float* x1      = thresh + 64;                              // B*C1*NP1
  float* x2      = x1 + (size_t)BATCH * C1 * NP1;            // B*C2*NP2

  float* sparse  = (float*)d_out;                            // B*2*EMBED
  float* dense   = sparse + (size_t)BATCH * 2 * EMBED;       // B*EMBED*NP2

  k_diff_stats<<<BATCH, 1024, 0, stream>>>(image, mirror, diff, thresh);
  k_centroid_sparse<<<BATCH, 256, 0, stream>>>(diff, thresh, gauss, nap, pos, sparse);
  k_conv1<<<(BATCH * NP1) / 256, 256, 0, stream>>>(diff, thresh, w1, b1, lw1, lb1, x1);
  k_conv2<<<(BATCH * NP2) / 256, 256, 0, stream>>>(x1, w2, b2, lw2, lb2, x2);
  // 64 imgs * 4096 tiles = 262144 waves; 8 waves/block -> 32768 blocks (exact, EXEC all-1s)
  k_conv3_wmma<<<32768, 256, 0, stream>>>(x2, w3, b3, dense);
}